// LeftRightMaxPooling_49452253446766
// MI455X (gfx1250) — compile-verified
//
#include <hip/hip_runtime.h>
#include <math.h>

// Row-parallel cummax (prefix max) for x[8,1,1024,4096] fp32.
// 8192 independent rows of 4096 floats; one wave32 per row.
// Bandwidth-bound: 256 MiB total traffic -> ~11.5 us floor at 23.3 TB/s.

typedef float v4f __attribute__((ext_vector_type(4)));

#define ROW_W 4096
#define ELEMS_PER_LANE 8            // two b128 loads per lane per chunk
#define CHUNK (32 * ELEMS_PER_LANE) // 256 elements per wave per chunk
#define NCHUNK (ROW_W / CHUNK)      // 16 chunks per row
#define WAVES_PER_BLOCK 8

__global__ __launch_bounds__(256, 4)
void cummax_rows_kernel(const float* __restrict__ x,
                        float* __restrict__ out,
                        long long n_rows) {
    const int lane = threadIdx.x & 31;
    const int wave = threadIdx.x >> 5;
    const long long row = (long long)blockIdx.x * WAVES_PER_BLOCK + wave;
    if (row >= n_rows) return;

    const float* __restrict__ src = x + row * (long long)ROW_W;
    float* __restrict__ dst       = out + row * (long long)ROW_W;

    const int lofs = lane * ELEMS_PER_LANE;

    float carry = -INFINITY;

    // Prime the pipeline: loads for chunk 0.
    v4f a = __builtin_nontemporal_load((const v4f*)(src + lofs));
    v4f b = __builtin_nontemporal_load((const v4f*)(src + lofs + 4));

    #pragma unroll 1
    for (int c = 0; c < NCHUNK; ++c) {
        // Issue next chunk's loads before this chunk's scan chain so the
        // VMEM latency overlaps the ALU/DS work (carry is the only serial dep).
        v4f na, nb;
        if (c + 1 < NCHUNK) {
            const float* nbase = src + (c + 1) * CHUNK + lofs;
            na = __builtin_nontemporal_load((const v4f*)nbase);
            nb = __builtin_nontemporal_load((const v4f*)(nbase + 4));
            // Keep the HBM stream primed two chunks ahead
            // (lowers to global_prefetch_b8 on gfx1250).
            if (c + 2 < NCHUNK) {
                __builtin_prefetch(nbase + CHUNK, 0, 0);
            }
        }

        // Serial prefix-max over the 8 elements this lane owns.
        a.y = fmaxf(a.y, a.x);
        a.z = fmaxf(a.z, a.y);
        a.w = fmaxf(a.w, a.z);
        b.x = fmaxf(b.x, a.w);
        b.y = fmaxf(b.y, b.x);
        b.z = fmaxf(b.z, b.y);
        b.w = fmaxf(b.w, b.z);

        // Wave32 inclusive max-scan of per-lane maxima (5 shuffle steps).
        float m = b.w;
        #pragma unroll
        for (int off = 1; off < 32; off <<= 1) {
            float t = __shfl_up(m, off, 32);
            if (lane >= off) m = fmaxf(m, t);
        }

        // Exclusive prefix for this lane, folded with the running carry
        // (prefix-max is monotone, so one fmax covers the whole chunk).
        float pre = __shfl_up(m, 1, 32);
        pre = (lane == 0) ? carry : fmaxf(pre, carry);

        a.x = fmaxf(a.x, pre); a.y = fmaxf(a.y, pre);
        a.z = fmaxf(a.z, pre); a.w = fmaxf(a.w, pre);
        b.x = fmaxf(b.x, pre); b.y = fmaxf(b.y, pre);
        b.z = fmaxf(b.z, pre); b.w = fmaxf(b.w, pre);

        float* obase = dst + c * CHUNK + lofs;
        __builtin_nontemporal_store(a, (v4f*)obase);
        __builtin_nontemporal_store(b, (v4f*)(obase + 4));

        // Wave-uniform carry: readlane(31) -> SGPR broadcast (no ds op).
        carry = __builtin_amdgcn_readlane(m, 31);

        a = na;
        b = nb;
    }
}

extern "C" void kernel_launch(void* const* d_in, const int* in_sizes, int n_in,
                              void* d_out, int out_size, void* d_ws, size_t ws_size,
                              hipStream_t stream) {
    const float* x = (const float*)d_in[0];
    float* out = (float*)d_out;

    const long long n = (long long)in_sizes[0];   // 8*1*1024*4096
    const long long n_rows = n / ROW_W;           // 8192

    const int blocks = (int)((n_rows + WAVES_PER_BLOCK - 1) / WAVES_PER_BLOCK);
    cummax_rows_kernel<<<blocks, 32 * WAVES_PER_BLOCK, 0, stream>>>(x, out, n_rows);
}